// AdaptiveOutputHead_17927193493834
// MI455X (gfx1250) — compile-verified
//
#include <hip/hip_runtime.h>
#include <hip/hip_bf16.h>
#include <math.h>

// Sampled-softmax loss for MI455X (gfx1250, wave32).
// V=128000, D=512, B=2, T=1024, K=512  ->  N = 2048 tokens.
//
// Strategy: one workgroup (256 thr = 8 waves) per token. Each wave handles
// 64 candidates as 4 WMMA tiles of 16 gathered weight rows, accumulating
// logits with V_WMMA_F32_16X16X4_F32 (f32-exact; compute is >100x under the
// memory roofline, so wasting WMMA columns is free). B-operand columns are
// replicated with h, so accumulator VGPRs hold logits directly.

typedef __attribute__((ext_vector_type(2))) float v2f;
typedef __attribute__((ext_vector_type(8))) float v8f;

#define D_DIM 512
#define K_CAND 512
#define NEG_PER 511

__global__ __launch_bounds__(256) void sampled_softmax_logits_kernel(
    const float* __restrict__ H,        // [N, 512]
    const int*   __restrict__ labels,   // [N]
    const float* __restrict__ W,        // [V, 512]
    const float* __restrict__ sampq,    // [V]
    const int*   __restrict__ negs,     // [N, 511]
    float*       __restrict__ loss_out) // [N]
{
    __shared__ float sh_h[D_DIM];
    __shared__ float sh_logits[K_CAND];
    __shared__ float red[256];

    const int n    = blockIdx.x;
    const int tid  = threadIdx.x;
    const int lane = tid & 31;
    const int wave = tid >> 5;      // 0..7
    const int m    = lane & 15;     // candidate row within tile
    const int half = lane >> 4;     // 0: K=0,1 side ; 1: K=2,3 side
    const int hoff = half * 4;

    // Stage h[n] into LDS: 256 threads x 2 floats.
    {
        const float2* hp = (const float2*)(H + (size_t)n * D_DIM);
        float2 t = hp[tid];
        sh_h[2 * tid + 0] = t.x;
        sh_h[2 * tid + 1] = t.y;
    }
    __syncthreads();

    // Gather this lane's 4 candidate row pointers (tiles of 16 rows).
    const float* rows[4];
#pragma unroll
    for (int t = 0; t < 4; ++t) {
        int c   = wave * 64 + t * 16 + m;                 // 0..511
        int idx = (c == 0) ? labels[n]
                           : negs[(size_t)n * NEG_PER + (c - 1)];
        rows[t] = W + (size_t)idx * D_DIM;
    }

    v8f acc[4];
#pragma unroll
    for (int t = 0; t < 4; ++t) acc[t] = 0.0f;            // vector splat

    // D-reduction: 64 iters x (1 ds_load_b128 + 4 global_load_b128 + 8 wmma).
    // Lane pair (l, l+16) covers one contiguous 32B span of a row.
    // The same half->K permutation is applied to A (weights) and B (h), so
    // the accumulated dot product is correct for either hardware assignment.
#pragma unroll 4
    for (int d0 = 0; d0 < D_DIM; d0 += 8) {
        float4 hv = *(const float4*)(sh_h + d0 + hoff);
        v2f b0 = {hv.x, hv.y};
        v2f b1 = {hv.z, hv.w};
#pragma unroll
        for (int t = 0; t < 4; ++t) {
            float4 wv = *(const float4*)(rows[t] + d0 + hoff);
            v2f a0 = {wv.x, wv.y};
            v2f a1 = {wv.z, wv.w};
            acc[t] = __builtin_amdgcn_wmma_f32_16x16x4_f32(
                false, a0, false, b0, (short)0, acc[t], false, false);
            acc[t] = __builtin_amdgcn_wmma_f32_16x16x4_f32(
                false, a1, false, b1, (short)0, acc[t], false, false);
        }
    }

    // C/D layout (16x16 f32): VGPR j -> M=j (lanes 0-15), M=j+8 (lanes 16-31);
    // all N columns identical (B replicated), so lane 0 / lane 16 publish.
    if (m == 0) {
        int mbase = wave * 64 + half * 8;
#pragma unroll
        for (int t = 0; t < 4; ++t)
#pragma unroll
            for (int j = 0; j < 8; ++j)
                sh_logits[mbase + t * 16 + j] = acc[t][j];
    }
    __syncthreads();

    // Stable logsumexp over the 512 logits (block tree reductions).
    float l1 = sh_logits[tid];
    float l2 = sh_logits[tid + 256];
    red[tid] = fmaxf(l1, l2);
    __syncthreads();
#pragma unroll
    for (int off = 128; off > 0; off >>= 1) {
        if (tid < off) red[tid] = fmaxf(red[tid], red[tid + off]);
        __syncthreads();
    }
    float gmax = red[0];
    __syncthreads();

    red[tid] = __expf(l1 - gmax) + __expf(l2 - gmax);
    __syncthreads();
#pragma unroll
    for (int off = 128; off > 0; off >>= 1) {
        if (tid < off) red[tid] += red[tid + off];
        __syncthreads();
    }

    if (tid == 0) {
        float lse    = gmax + __logf(red[0]);
        float q      = fmaxf(sampq[labels[n]], 1e-10f);
        float logit0 = sh_logits[0];               // true-label column
        loss_out[n]  = -logit0 + lse + __logf(q);
    }
}

// Deterministic mean over the per-token losses (no float atomics).
__global__ __launch_bounds__(256) void reduce_mean_kernel(
    const float* __restrict__ losses, float* __restrict__ out, int n)
{
    __shared__ float red[256];
    float s = 0.0f;
    for (int i = threadIdx.x; i < n; i += 256) s += losses[i];
    red[threadIdx.x] = s;
    __syncthreads();
#pragma unroll
    for (int off = 128; off > 0; off >>= 1) {
        if (threadIdx.x < off) red[threadIdx.x] += red[threadIdx.x + off];
        __syncthreads();
    }
    if (threadIdx.x == 0) out[0] = red[0] / (float)n;
}

extern "C" void kernel_launch(void* const* d_in, const int* in_sizes, int n_in,
                              void* d_out, int out_size, void* d_ws, size_t ws_size,
                              hipStream_t stream) {
    const float* H      = (const float*)d_in[0];  // hidden_states [B,T,D]
    const int*   labels = (const int*)  d_in[1];  // [B,T]
    const float* W      = (const float*)d_in[2];  // weight [V,D]
    const float* sampq  = (const float*)d_in[3];  // sampling_probs [V]
    const int*   negs   = (const int*)  d_in[4];  // neg_samples [N, K-1]

    const int N = in_sizes[1];                    // B*T = 2048
    float* loss_ws = (float*)d_ws;                // N floats of scratch
    float* out     = (float*)d_out;               // scalar mean loss (f32)

    sampled_softmax_logits_kernel<<<N, 256, 0, stream>>>(
        H, labels, W, sampq, negs, loss_ws);
    reduce_mean_kernel<<<1, 256, 0, stream>>>(loss_ws, out, N);
}